// TrainableQuanvLayer_19104014532608
// MI455X (gfx1250) — compile-verified
//
#include <hip/hip_runtime.h>
#include <math.h>

typedef __attribute__((ext_vector_type(2))) float v2f;
typedef __attribute__((ext_vector_type(8))) float v8f;

#define NB 16
#define NC 3
#define IH 224
#define IW 224
#define OH 112
#define OW 112
#define TPR (OW / 16)  /* 7 tiles per output row */

struct cplx { float re, im; };

// ---------------------------------------------------------------------------
// Setup: build per-layer shared 16x16 complex matrices M_l from weights.
// Thread t: layer = t>>4, evolves basis column (t&15).
// ws layout: float M[2][3][16][16], part 0 = re, 1 = im, 2 = -im.
// ---------------------------------------------------------------------------
__global__ void quanv_setup(const float* __restrict__ w, float* __restrict__ M)
{
    int t = threadIdx.x;
    if (t >= 32) return;
    int layer = t >> 4;
    int col   = t & 15;

    cplx st[16];
#pragma unroll
    for (int i = 0; i < 16; ++i) st[i] = { (i == col) ? 1.f : 0.f, 0.f };

    // RY(w[l][q][0]) then RZ(w[l][q][1]) per qubit q
    for (int q = 0; q < 4; ++q) {
        float ty = w[(layer * 4 + q) * 3 + 0] * 0.5f;
        float cy = cosf(ty), sy = sinf(ty);
        int m = 3 - q, mask = 1 << m;
#pragma unroll
        for (int i = 0; i < 16; ++i) if (!(i & mask)) {
            cplx a = st[i], b = st[i | mask];
            st[i]        = { cy * a.re - sy * b.re, cy * a.im - sy * b.im };
            st[i | mask] = { sy * a.re + cy * b.re, sy * a.im + cy * b.im };
        }
        float tz = w[(layer * 4 + q) * 3 + 1] * 0.5f;
        float cz = cosf(tz), sz = sinf(tz);
        // e = exp(-i*tz); bit=0 gets e, bit=1 gets conj(e)
#pragma unroll
        for (int i = 0; i < 16; ++i) {
            float er = cz, ei = ((i >> m) & 1) ? sz : -sz;
            cplx a = st[i];
            st[i] = { er * a.re - ei * a.im, er * a.im + ei * a.re };
        }
    }
    // CNOT ring permutations (0,1),(1,2),(2,3),(3,0): new[i] = old[perm[i]]
    const int cc[4] = {0, 1, 2, 3}, tq[4] = {1, 2, 3, 0};
    for (int p = 0; p < 4; ++p) {
        cplx ns[16];
#pragma unroll
        for (int i = 0; i < 16; ++i) {
            int ctrl = (i >> (3 - cc[p])) & 1;
            ns[i] = st[i ^ (ctrl << (3 - tq[p]))];
        }
#pragma unroll
        for (int i = 0; i < 16; ++i) st[i] = ns[i];
    }
    // final RY(w[l][q][2])
    for (int q = 0; q < 4; ++q) {
        float ty = w[(layer * 4 + q) * 3 + 2] * 0.5f;
        float cy = cosf(ty), sy = sinf(ty);
        int m = 3 - q, mask = 1 << m;
#pragma unroll
        for (int i = 0; i < 16; ++i) if (!(i & mask)) {
            cplx a = st[i], b = st[i | mask];
            st[i]        = { cy * a.re - sy * b.re, cy * a.im - sy * b.im };
            st[i | mask] = { sy * a.re + cy * b.re, sy * a.im + cy * b.im };
        }
    }
#pragma unroll
    for (int r = 0; r < 16; ++r) {
        M[((layer * 3 + 0) * 16 + r) * 16 + col] =  st[r].re;
        M[((layer * 3 + 1) * 16 + r) * 16 + col] =  st[r].im;
        M[((layer * 3 + 2) * 16 + r) * 16 + col] = -st[r].im;
    }
}

// Half-wave exchange (lanes l <-> l^16): ds_swizzle SWAPX16, zero VALU setup.
__device__ __forceinline__ float swapx16(float x)
{
    return __int_as_float(__builtin_amdgcn_ds_swizzle(__float_as_int(x), 0x401F));
}

// Complex 16x16 matvec batched over 16 patches via 16 chained f32 WMMAs.
// Cre = Mre*Bre + (-Mim)*Bim ; Cim = Mim*Bre + Mre*Bim
__device__ __forceinline__ void cmatvec(const v2f Are[4], const v2f Aim[4],
                                        const v2f Aimn[4],
                                        const v2f Bre[4], const v2f Bim[4],
                                        v8f& Cre, v8f& Cim)
{
    Cre = (v8f){0.f, 0.f, 0.f, 0.f, 0.f, 0.f, 0.f, 0.f};
    Cim = (v8f){0.f, 0.f, 0.f, 0.f, 0.f, 0.f, 0.f, 0.f};
#pragma unroll
    for (int j = 0; j < 4; ++j)
        Cre = __builtin_amdgcn_wmma_f32_16x16x4_f32(false, Are[j],  false, Bre[j], (short)0, Cre, false, false);
#pragma unroll
    for (int j = 0; j < 4; ++j)
        Cre = __builtin_amdgcn_wmma_f32_16x16x4_f32(false, Aimn[j], false, Bim[j], (short)0, Cre, false, false);
#pragma unroll
    for (int j = 0; j < 4; ++j)
        Cim = __builtin_amdgcn_wmma_f32_16x16x4_f32(false, Aim[j],  false, Bre[j], (short)0, Cim, false, false);
#pragma unroll
    for (int j = 0; j < 4; ++j)
        Cim = __builtin_amdgcn_wmma_f32_16x16x4_f32(false, Are[j],  false, Bim[j], (short)0, Cim, false, false);
}

// ---------------------------------------------------------------------------
// Main: one wave = one tile of 16 patches (16 consecutive ow).
// Grid: (98, 48) x 256 threads. blockIdx.y = b*NC+c; blockIdx.x*8+wid = tile
// index within the (b,c) image (784 = 112*7 tiles).
// C/D layout: lane l holds amplitudes i = v + 8*(l>=16), v=0..7, patch (l&15).
// B layout:   lane l holds k = 4j + 2*(l>=16) + e, chunk j, patch (l&15).
// ---------------------------------------------------------------------------
__global__ void __launch_bounds__(256) quanv_main(const float* __restrict__ x,
                                                  const float* __restrict__ M,
                                                  const float* __restrict__ gscale,
                                                  float* __restrict__ out)
{
    const int  lane = threadIdx.x & 31;
    const bool hi   = lane >= 16;
    const int  r    = lane & 15;

    const int widx = blockIdx.x * 8 + (threadIdx.x >> 5); // 0..783
    const int oh   = widx / TPR;
    const int owb  = widx - oh * TPR;
    const int bc   = blockIdx.y;                          // b*NC + c
    const int ow   = owb * 16 + r;

    // WMMA A-fragments of M1/M2.
    // A 16x4 f32 layout: lanes 0-15 -> (row=lane, K=0,1); lanes 16-31 -> (row, K=2,3)
    v2f A[2][3][4];
#pragma unroll
    for (int l = 0; l < 2; ++l)
#pragma unroll
        for (int p = 0; p < 3; ++p)
#pragma unroll
            for (int j = 0; j < 4; ++j) {
                int k0 = 4 * j + (hi ? 2 : 0);
                const float* src = &M[((l * 3 + p) * 16 + r) * 16 + k0];
                v2f a; a[0] = src[0]; a[1] = src[1];
                A[l][p][j] = a;
            }
    const float gs = gscale[0];

    // Patch angles (both half-waves read the same patch — L0 hit).
    const float* row0 = x + (((size_t)bc * IH + 2 * oh) * IW) + 2 * ow;
    const float2 t01 = *(const float2*)row0;
    const float2 t23 = *(const float2*)(row0 + IW);
    float th[4] = { t01.x, t01.y, t23.x, t23.y };
    float cq[4], sq[4];
#pragma unroll
    for (int q = 0; q < 4; ++q) { cq[q] = __cosf(th[q] * 0.5f); sq[q] = __sinf(th[q] * 0.5f); }

    // psi0 = RX⊗4 |0>: amplitude k = (-i)^popc(k) * prod(bit? s : c).
    // k = 4j + 2*hi + e: bits [3:2]=j (qubits 0,1), bit1=hi (qubit 2), bit0=e (qubit 3)
    // => only the qubit-2 factor and a popcount increment are lane-dependent.
    const float f2  = hi ? sq[2] : cq[2];
    const float fe0 = f2 * cq[3], fe1 = f2 * sq[3];
    v2f Bre[4], Bim[4];
#pragma unroll
    for (int j = 0; j < 4; ++j) {
        const float pj = ((j >> 1) ? sq[0] : cq[0]) * ((j & 1) ? sq[1] : cq[1]);
#pragma unroll
        for (int e = 0; e < 2; ++e) {
            const float amp  = pj * (e ? fe1 : fe0);
            const int   base = (j >> 1) + (j & 1) + e;          // compile-time
            // unit (-i)^base placement at compile time, then *(-i) if hi:
            const float re0 = (base == 0) ? amp : (base == 2) ? -amp : 0.f;
            const float im0 = (base == 1) ? -amp : (base == 3) ? amp : 0.f;
            Bre[j][e] = hi ? im0 : re0;
            Bim[j][e] = hi ? -re0 : im0;
        }
    }

    // psi1 = M1 * psi0
    v8f Cre, Cim;
    cmatvec(A[0][0], A[0][1], A[0][2], Bre, Bim, Cre, Cim);

    // Mid-circuit RX⊗4, qubits 1..3 (state bits 2,1,0) in C/D layout, in-register.
    // RX is symmetric: new = c*x + (-i*s)*partner.
    float Re[8], Im[8];
#pragma unroll
    for (int v = 0; v < 8; ++v) { Re[v] = Cre[v]; Im[v] = Cim[v]; }
#pragma unroll
    for (int q = 1; q < 4; ++q) {
        int mask = 1 << (3 - q);
        float nRe[8], nIm[8];
#pragma unroll
        for (int v = 0; v < 8; ++v) {
            int u = v ^ mask;
            nRe[v] = cq[q] * Re[v] + sq[q] * Im[u];
            nIm[v] = cq[q] * Im[v] - sq[q] * Re[u];
        }
#pragma unroll
        for (int v = 0; v < 8; ++v) { Re[v] = nRe[v]; Im[v] = nIm[v]; }
    }

    // Relayout C/D -> B (one SWAPX16 round: 8 re + 8 im swizzles).
    float pRe[8], pIm[8];
#pragma unroll
    for (int v = 0; v < 8; ++v) { pRe[v] = swapx16(Re[v]); pIm[v] = swapx16(Im[v]); }
    v2f B2re[4], B2im[4];
    // lo lanes need k = 4j+{0,1}; hi lanes need k = 4j+{2,3}
    B2re[0][0] = hi ? pRe[2] : Re[0];  B2re[0][1] = hi ? pRe[3] : Re[1];
    B2re[1][0] = hi ? pRe[6] : Re[4];  B2re[1][1] = hi ? pRe[7] : Re[5];
    B2re[2][0] = hi ? Re[2] : pRe[0];  B2re[2][1] = hi ? Re[3] : pRe[1];
    B2re[3][0] = hi ? Re[6] : pRe[4];  B2re[3][1] = hi ? Re[7] : pRe[5];
    B2im[0][0] = hi ? pIm[2] : Im[0];  B2im[0][1] = hi ? pIm[3] : Im[1];
    B2im[1][0] = hi ? pIm[6] : Im[4];  B2im[1][1] = hi ? pIm[7] : Im[5];
    B2im[2][0] = hi ? Im[2] : pIm[0];  B2im[2][1] = hi ? Im[3] : pIm[1];
    B2im[3][0] = hi ? Im[6] : pIm[4];  B2im[3][1] = hi ? Im[7] : pIm[5];

    // Qubit-0 RX (state bit 3) in B layout: bit3(k) == (j>=2), so the pair
    // (k, k^8) = chunks (j, j^2) lives in the SAME lane -> in-register.
#pragma unroll
    for (int e = 0; e < 2; ++e) {
        float r0 = B2re[0][e], r1 = B2re[1][e], r2 = B2re[2][e], r3 = B2re[3][e];
        float i0 = B2im[0][e], i1 = B2im[1][e], i2 = B2im[2][e], i3 = B2im[3][e];
        B2re[0][e] = cq[0] * r0 + sq[0] * i2;  B2im[0][e] = cq[0] * i0 - sq[0] * r2;
        B2re[1][e] = cq[0] * r1 + sq[0] * i3;  B2im[1][e] = cq[0] * i1 - sq[0] * r3;
        B2re[2][e] = cq[0] * r2 + sq[0] * i0;  B2im[2][e] = cq[0] * i2 - sq[0] * r0;
        B2re[3][e] = cq[0] * r3 + sq[0] * i1;  B2im[3][e] = cq[0] * i3 - sq[0] * r1;
    }

    // psi2 = M2 * psi1'
    v8f Dre, Dim;
    cmatvec(A[1][0], A[1][1], A[1][2], B2re, B2im, Dre, Dim);

    // probs + Z expectations; each lane holds i = v + 8*hi of its patch.
    float part[4] = {0.f, 0.f, 0.f, 0.f};
#pragma unroll
    for (int v = 0; v < 8; ++v) {
        int i = v + (hi ? 8 : 0);
        float p = Dre[v] * Dre[v] + Dim[v] * Dim[v];
#pragma unroll
        for (int q = 0; q < 4; ++q)
            part[q] += (((i >> (3 - q)) & 1) ? -p : p);
    }
#pragma unroll
    for (int q = 0; q < 4; ++q) part[q] += swapx16(part[q]);

    // out[b][c*4+q][oh][ow]; lo lanes store q=0,1; hi lanes store q=2,3.
    int qb = hi ? 2 : 0;
    size_t obase = ((size_t)(bc * 4) * OH + oh) * OW + ow;
#pragma unroll
    for (int e = 0; e < 2; ++e) {
        int q = qb + e;
        out[obase + (size_t)q * OH * OW] = part[q] * gs;
    }
}

extern "C" void kernel_launch(void* const* d_in, const int* in_sizes, int n_in,
                              void* d_out, int out_size, void* d_ws, size_t ws_size,
                              hipStream_t stream)
{
    const float* x  = (const float*)d_in[0];
    const float* w  = (const float*)d_in[1];
    const float* gs = (const float*)d_in[2];
    float* out = (float*)d_out;
    float* M   = (float*)d_ws;   // 2*3*16*16 floats = 6 KB

    quanv_setup<<<1, 32, 0, stream>>>(w, M);
    quanv_main<<<dim3(98, 48), 256, 0, stream>>>(x, M, gs, out);
    (void)in_sizes; (void)n_in; (void)out_size; (void)ws_size;
}